// TAG_19430432047388
// MI455X (gfx1250) — compile-verified
//
#include <hip/hip_runtime.h>

// GCN on gfx1250: fp32 WMMA (16x16x4) GEMMs + L2-resident atomic scatter.

typedef __attribute__((ext_vector_type(2))) float v2f;
typedef __attribute__((ext_vector_type(8))) float v8f;

#define HD 64            // feature / hidden dim (N_FEAT == H_DIM == 64)

// ---------------------------------------------------------------- degree ---
__global__ void k_deg_init(float* __restrict__ deg, int n) {
  int i = blockIdx.x * blockDim.x + threadIdx.x;
  if (i < n) deg[i] = 1.0f;                 // self-loop contribution
}

__global__ void k_deg_scatter(const int* __restrict__ col, float* __restrict__ deg, int e) {
  int i = blockIdx.x * blockDim.x + threadIdx.x;
  if (i < e) atomicAdd(&deg[col[i]], 1.0f);
}

__global__ void k_deg_final(float* __restrict__ deg, int n) {
  int i = blockIdx.x * blockDim.x + threadIdx.x;
  if (i < n) {
    float d = deg[i];
    deg[i] = (d > 0.0f) ? rsqrtf(d) : 0.0f;  // deg is now dinv
  }
}

// ------------------------------------------------------------ WMMA GEMM ----
// hout[N,64] = hin[N,64] @ W[64,64]  (+ bias if non-null)
// Block = 128 threads = 4 waves; each wave computes a 16x64 tile.
__global__ __launch_bounds__(128)
void k_gemm64_wmma(const float* __restrict__ hin, const float* __restrict__ W,
                   const float* __restrict__ bias, float* __restrict__ hout, int n) {
  __shared__ float sW[HD * HD];            // 16 KB of the 320 KB WGP LDS

  const int tid = threadIdx.x;
  {   // cooperative 64x64 W load: 1024 float4, 8 per thread
    const float4* src = (const float4*)W;
    float4*       dst = (float4*)sW;
#pragma unroll
    for (int i = 0; i < 8; ++i) dst[tid + i * 128] = src[tid + i * 128];
  }
  __syncthreads();

  const int wave    = tid >> 5;
  const int lane    = tid & 31;
  const int half    = lane >> 4;           // 0: lanes 0-15, 1: lanes 16-31
  const int l16     = lane & 15;
  const int rowBase = blockIdx.x * 64 + wave * 16;
  if (rowBase >= n) return;                // N % 16 == 0 -> whole-wave tiles only

  const v8f vzero = {0.f, 0.f, 0.f, 0.f, 0.f, 0.f, 0.f, 0.f};
  v8f acc[4] = {vzero, vzero, vzero, vzero};

  const float* arow = hin + (size_t)(rowBase + l16) * HD;

#pragma unroll
  for (int kb = 0; kb < HD; kb += 4) {
    // A 16x4 f32: VGPR v holds K = kb + 2*half + v  -> contiguous pair
    v2f a = *(const v2f*)(arow + kb + 2 * half);
#pragma unroll
    for (int nt = 0; nt < 4; ++nt) {
      const int c = nt * 16 + l16;
      // B 4x16 f32: VGPR v holds K = kb + 2*half + v, col = c
      v2f b;
      b.x = sW[(kb + 2 * half + 0) * HD + c];
      b.y = sW[(kb + 2 * half + 1) * HD + c];
      acc[nt] = __builtin_amdgcn_wmma_f32_16x16x4_f32(
          /*neg_a=*/false, a, /*neg_b=*/false, b,
          /*c_mod=*/(short)0, acc[nt], /*reuse_a=*/false, /*reuse_b=*/false);
    }
  }

  // D 16x16 f32: VGPR v is row (v + 8*half), col l16
#pragma unroll
  for (int nt = 0; nt < 4; ++nt) {
    const int c = nt * 16 + l16;
    const float badd = bias ? bias[c] : 0.0f;
#pragma unroll
    for (int v = 0; v < 8; ++v) {
      const int r = rowBase + v + 8 * half;
      hout[(size_t)r * HD + c] = acc[nt][v] + badd;
    }
  }
}

// --------------------------------------------------------- aggregation -----
// agg[i,:] = h1[i,:] * dinv[i]^2   (self-loop edge, also zero-initializes agg)
__global__ void k_agg_init(float* __restrict__ agg, const float* __restrict__ h1,
                           const float* __restrict__ dinv, int n) {
  int i = blockIdx.x * blockDim.x + threadIdx.x;  // over n*16 float4s
  if (i < n * (HD / 4)) {
    int node = i >> 4;
    float s = dinv[node];
    s = s * s;
    float4 v = ((const float4*)h1)[i];
    v.x *= s; v.y *= s; v.z *= s; v.w *= s;
    ((float4*)agg)[i] = v;
  }
}

// agg[col,:] += h1[row,:] * dinv[row]*dinv[col]; one wave per edge, float2/lane
__global__ __launch_bounds__(256)
void k_edge_scatter(float* __restrict__ agg, const float* __restrict__ h1,
                    const int* __restrict__ ei, const float* __restrict__ dinv, int e) {
  int gid  = blockIdx.x * blockDim.x + threadIdx.x;
  int edge = gid >> 5;
  if (edge >= e) return;
  int lane = gid & 31;
  int r = ei[edge];        // row = source
  int c = ei[e + edge];    // col = destination
  float s = dinv[r] * dinv[c];
  float2 v = ((const float2*)(h1 + (size_t)r * HD))[lane];
  float* dst = agg + (size_t)c * HD + lane * 2;
  atomicAdd(dst + 0, v.x * s);
  atomicAdd(dst + 1, v.y * s);
}

// dst = relu(src + bias)
__global__ void k_bias_relu(float* __restrict__ dst, const float* __restrict__ src,
                            const float* __restrict__ bias, int n) {
  int i = blockIdx.x * blockDim.x + threadIdx.x;  // over n*64 floats
  if (i < n * HD) {
    float v = src[i] + bias[i & (HD - 1)];
    dst[i] = v > 0.0f ? v : 0.0f;
  }
}

// ------------------------------------------------------------- driver ------
extern "C" void kernel_launch(void* const* d_in, const int* in_sizes, int n_in,
                              void* d_out, int out_size, void* d_ws, size_t ws_size,
                              hipStream_t stream) {
  const float* x  = (const float*)d_in[0];
  const int*   ei = (const int*)d_in[1];   // [2, E] (row; col)
  // d_in[2] = batch : unused by the reference outputs
  const float* W0 = (const float*)d_in[3];
  const float* b0 = (const float*)d_in[4];
  const float* W1 = (const float*)d_in[5];
  const float* b1 = (const float*)d_in[6];
  const float* W2 = (const float*)d_in[7];
  const float* b2 = (const float*)d_in[8];
  const float* Wl = (const float*)d_in[9];
  const float* bl = (const float*)d_in[10];

  const int N = in_sizes[0] / HD;
  const int E = in_sizes[1] / 2;

  float* out    = (float*)d_out;           // [0, N*64) hidden, [N*64, 2N*64) logits
  float* logits = out + (size_t)N * HD;

  // workspace: dinv[N] | A[N*64] | B[N*64]
  float* dinv = (float*)d_ws;
  size_t off  = ((size_t)N + 255) & ~(size_t)255;
  float* A    = dinv + off;
  float* B    = A + (size_t)N * HD;

  const int TB = 256;
  dim3 blkN((N + TB - 1) / TB);
  dim3 blkE((E + TB - 1) / TB);
  dim3 blkF4(((size_t)N * (HD / 4) + TB - 1) / TB);   // per-float4 elementwise
  dim3 blkF (((size_t)N * HD + TB - 1) / TB);         // per-float elementwise
  dim3 blkSc(((size_t)E * 32 + TB - 1) / TB);         // one wave per edge
  dim3 blkG ((N + 63) / 64);                          // GEMM: 64 rows / block

  // degrees -> dinv
  k_deg_init   <<<blkN, TB, 0, stream>>>(dinv, N);
  k_deg_scatter<<<blkE, TB, 0, stream>>>(ei + E, dinv, E);
  k_deg_final  <<<blkN, TB, 0, stream>>>(dinv, N);

  // layer 0: x -> A
  k_gemm64_wmma <<<blkG, 128, 0, stream>>>(x, W0, nullptr, B, N);
  k_agg_init    <<<blkF4, TB, 0, stream>>>(A, B, dinv, N);
  k_edge_scatter<<<blkSc, TB, 0, stream>>>(A, B, ei, dinv, E);
  k_bias_relu   <<<blkF,  TB, 0, stream>>>(A, A, b0, N);

  // layer 1: A -> A
  k_gemm64_wmma <<<blkG, 128, 0, stream>>>(A, W1, nullptr, B, N);
  k_agg_init    <<<blkF4, TB, 0, stream>>>(A, B, dinv, N);
  k_edge_scatter<<<blkSc, TB, 0, stream>>>(A, B, ei, dinv, E);
  k_bias_relu   <<<blkF,  TB, 0, stream>>>(A, A, b1, N);

  // layer 2: A -> out (first output)
  k_gemm64_wmma <<<blkG, 128, 0, stream>>>(A, W2, nullptr, B, N);
  k_agg_init    <<<blkF4, TB, 0, stream>>>(A, B, dinv, N);
  k_edge_scatter<<<blkSc, TB, 0, stream>>>(A, B, ei, dinv, E);
  k_bias_relu   <<<blkF,  TB, 0, stream>>>(out, A, b2, N);

  // head: logits = out @ Wl + bl (second output)
  k_gemm64_wmma <<<blkG, 128, 0, stream>>>(out, Wl, bl, logits, N);
}